// GNN_BranchModel_70935679861201
// MI455X (gfx1250) — compile-verified
//
#include <hip/hip_runtime.h>

// Problem constants (fixed by the reference's setup_inputs)
#define BSZ   64
#define NF    512
#define DIM   510
#define TOTAL 1022
#define MAXIT 50
#define TOLV  1e-5f

typedef float v2f __attribute__((ext_vector_type(2)));
typedef float v8f __attribute__((ext_vector_type(8)));

static constexpr long X_ELEMS    = (long)BSZ * DIM * NF;   // ping-pong X buffer in ws
static constexpr long PART_ELEMS = (long)BSZ * DIM;        // per-row diff partials

// ---------------------------------------------------------------------------
// Control init: anybad[0]=1 (iteration 1 always runs), anybad[1..50]=0.
// ---------------------------------------------------------------------------
__global__ void init_kernel(int* __restrict__ anybad, int* __restrict__ last_iter) {
  int t = threadIdx.x;
  if (t <= MAXIT) anybad[t] = (t == 0) ? 1 : 0;
  if (t == 0) *last_iter = 1;
}

// ---------------------------------------------------------------------------
// Leaf region of the output: out[b, j, :] = identity[j, :] for j < NF.
// One 2KB row per 128-thread block; b128 stores.
// ---------------------------------------------------------------------------
__global__ void leaf_kernel(const float* __restrict__ ident, float* __restrict__ out) {
  const int row = blockIdx.x;          // 0 .. BSZ*NF-1
  const int b   = row >> 9;            // /512
  const int j   = row & (NF - 1);
  const int t   = threadIdx.x;         // 0..127, one float4 each
  float4 v = ((const float4*)(ident + (long)j * NF))[t];
  ((float4*)(out + (long)b * TOTAL * NF + (long)j * NF))[t] = v;
}

// ---------------------------------------------------------------------------
// Wave-level sum of 32 lane partials using V_WMMA_F32_16X16X4_F32.
// A = ones(16x4); B rows 0..1 carry the 32 lane values, rows 2..3 zero.
// D[m,n] = p[n] + p[n+16]  -> finish with 4 shfl_xor steps. Deterministic.
// ---------------------------------------------------------------------------
__device__ __forceinline__ float wave_sum_wmma(float p) {
  v2f a;  a.x  = 1.0f; a.y  = 1.0f;    // A-matrix: all ones
  v2f bm; bm.x = p;    bm.y = 0.0f;    // B-matrix: rows 0..1 = partials
  v8f c = {};
  c = __builtin_amdgcn_wmma_f32_16x16x4_f32(false, a, false, bm, (short)0, c,
                                            false, false);
  float s = c[0];                      // s(lane L) = p[L&15] + p[(L&15)+16]
  s += __shfl_xor(s, 1, 32);
  s += __shfl_xor(s, 2, 32);
  s += __shfl_xor(s, 4, 32);
  s += __shfl_xor(s, 8, 32);
  return s;
}

// ---------------------------------------------------------------------------
// One fixpoint step for one output row (b,i):
//   dst[b,i,:] = (full[j0,:] + full[j1,:] + full[j2,:]) / 3
// where full rows < NF come from the 1MB identity matrix (cache resident)
// and rows >= NF from the previous X buffer (L2 resident, ~134MB live set
// fits the MI455X 192MB L2). Iteration 1 uses the constant X0 = 1/NF.
// Also emits the per-row L1-diff partial (deterministic, no atomics).
// ---------------------------------------------------------------------------
__global__ void step_kernel(const float* __restrict__ ident,
                            const int*  __restrict__ eidx,
                            const float* __restrict__ srcX, long srcStride,
                            float*       __restrict__ dstX, long dstStride,
                            float* __restrict__ partial,
                            const int* __restrict__ anybad,
                            int* __restrict__ last_iter,
                            int iter, int first)
{
  if (anybad[iter - 1] == 0) return;   // fixpoint already reached: skip
  const int row = blockIdx.x;          // 0 .. BSZ*DIM-1
  const int b   = row / DIM;
  const int i   = row - b * DIM;
  const int t   = threadIdx.x;         // 0..127, one float4 each
  if (row == 0 && t == 0) *last_iter = iter;

  const int* e = eidx + ((long)b * TOTAL + NF + i) * 3;
  const int j0 = e[0], j1 = e[1], j2 = e[2];

  float4 acc = make_float4(0.f, 0.f, 0.f, 0.f);
#pragma unroll
  for (int k = 0; k < 3; ++k) {
    const int j = (k == 0) ? j0 : (k == 1) ? j1 : j2;
    float4 v;
    if (j < NF) {
      v = ((const float4*)(ident + (long)j * NF))[t];
    } else if (first) {
      v = make_float4(1.f / NF, 1.f / NF, 1.f / NF, 1.f / NF);
    } else {
      v = ((const float4*)(srcX + (long)b * srcStride + (long)(j - NF) * NF))[t];
    }
    acc.x += v.x; acc.y += v.y; acc.z += v.z; acc.w += v.w;
  }
  float4 o;
  o.x = acc.x / 3.0f; o.y = acc.y / 3.0f; o.z = acc.z / 3.0f; o.w = acc.w / 3.0f;

  float4 prev;
  if (first) prev = make_float4(1.f / NF, 1.f / NF, 1.f / NF, 1.f / NF);
  else       prev = ((const float4*)(srcX + (long)b * srcStride + (long)i * NF))[t];

  ((float4*)(dstX + (long)b * dstStride + (long)i * NF))[t] = o;

  float p = fabsf(o.x - prev.x) + fabsf(o.y - prev.y) +
            fabsf(o.z - prev.z) + fabsf(o.w - prev.w);

  const float s = wave_sum_wmma(p);    // EXEC is all-ones here (uniform control flow)

  __shared__ float wsum[4];
  if ((t & 31) == 0) wsum[t >> 5] = s;
  __syncthreads();
  if (t == 0) partial[row] = (wsum[0] + wsum[1]) + (wsum[2] + wsum[3]);
}

// ---------------------------------------------------------------------------
// Per-batch convergence check: lnorm_b = sum(partial[b,:]) / (DIM*NF).
// Sticky gating: only evaluated while anybad[iter-1]!=0; sets anybad[iter]=1
// if any tree still moves. Fixed-order tree reduction -> deterministic.
// ---------------------------------------------------------------------------
__global__ void check_kernel(const float* __restrict__ partial,
                             int* __restrict__ anybad, int iter)
{
  if (anybad[iter - 1] == 0) return;   // converged: anybad[iter] stays 0
  const int b = blockIdx.x;            // 64 blocks, one per batch
  const int t = threadIdx.x;           // 128 threads
  float s = 0.f;
  for (int i = t; i < DIM; i += 128) s += partial[(long)b * DIM + i];
  __shared__ float red[128];
  red[t] = s;
  __syncthreads();
  for (int o = 64; o > 0; o >>= 1) {
    if (t < o) red[t] += red[t + o];
    __syncthreads();
  }
  if (t == 0) {
    const float lnorm = red[0] / (float)((long)DIM * NF);
    if (lnorm > TOLV) anybad[iter] = 1;
  }
}

// ---------------------------------------------------------------------------
// If the last executed iteration was even, its X lives in ws: copy to d_out.
// Odd iterations already wrote d_out's X region directly.
// ---------------------------------------------------------------------------
__global__ void finalize_kernel(const float* __restrict__ wsX,
                                float* __restrict__ out,
                                const int* __restrict__ last_iter)
{
  if ((*last_iter) & 1) return;        // final X already in d_out
  const int row = blockIdx.x;
  const int b   = row / DIM;
  const int i   = row - b * DIM;
  const int t   = threadIdx.x;
  ((float4*)(out + (long)b * TOTAL * NF + (long)(NF + i) * NF))[t] =
      ((const float4*)(wsX + (long)b * DIM * NF + (long)i * NF))[t];
}

// ---------------------------------------------------------------------------
// Launch: static graph of 1 init + 1 leaf + 50x(step[,check]) + 1 finalize.
// Data-dependent trip count is handled by device-side sticky flags so the
// same kernels are launched every call (graph-capture safe, deterministic).
// ---------------------------------------------------------------------------
extern "C" void kernel_launch(void* const* d_in, const int* in_sizes, int n_in,
                              void* d_out, int out_size, void* d_ws, size_t ws_size,
                              hipStream_t stream) {
  (void)in_sizes; (void)n_in; (void)out_size; (void)ws_size;

  const float* ident = (const float*)d_in[0];   // (512,512) f32
  const int*   eidx  = (const int*)d_in[1];     // (64,1022,3) i32
  float*       out   = (float*)d_out;           // (64,1022,512) f32

  float* wsX      = (float*)d_ws;               // even-iteration X buffer
  float* partial  = wsX + X_ELEMS;              // per-row diff partials
  int*   anybad   = (int*)(partial + PART_ELEMS);
  int*   last_it  = anybad + (MAXIT + 1);

  init_kernel<<<1, 64, 0, stream>>>(anybad, last_it);
  leaf_kernel<<<BSZ * NF, 128, 0, stream>>>(ident, out);

  float* bufA = out + (long)NF * NF;            // d_out X region (odd iters)
  const long strideA = (long)TOTAL * NF;
  float* bufB = wsX;                            // ws X buffer (even iters)
  const long strideB = (long)DIM * NF;

  for (int k = 1; k <= MAXIT; ++k) {
    const float* src; long ss; float* dst; long ds;
    if (k & 1) { src = bufB; ss = strideB; dst = bufA; ds = strideA; }
    else       { src = bufA; ss = strideA; dst = bufB; ds = strideB; }
    step_kernel<<<BSZ * DIM, 128, 0, stream>>>(ident, eidx, src, ss, dst, ds,
                                               partial, anybad, last_it,
                                               k, (k == 1) ? 1 : 0);
    if (k < MAXIT)
      check_kernel<<<BSZ, 128, 0, stream>>>(partial, anybad, k);
  }
  finalize_kernel<<<BSZ * DIM, 128, 0, stream>>>(bufB, out, last_it);
}